// BaseNet_44659069944275
// MI455X (gfx1250) — compile-verified
//
#include <hip/hip_runtime.h>

// GCN dims from the reference
#define F_IN  128
#define F_HID 64
#define F_OUT 128

typedef __attribute__((ext_vector_type(2))) float v2f;
typedef __attribute__((ext_vector_type(8))) float v8f;

// ---------------------------------------------------------------------------
// Degree: deg[dst] += 1 per edge (atomic scatter)
// ---------------------------------------------------------------------------
__global__ void deg_kernel(const int* __restrict__ dst, float* __restrict__ deg, int E) {
    int i = blockIdx.x * blockDim.x + threadIdx.x;
    if (i < E) atomicAdd(&deg[dst[i]], 1.0f);
}

// norm[i] = rsqrt(deg[i] + 1)  (in-place over the deg buffer)
__global__ void norm_kernel(float* __restrict__ norm, int N) {
    int i = blockIdx.x * blockDim.x + threadIdx.x;
    if (i < N) norm[i] = rsqrtf(norm[i] + 1.0f);
}

// ---------------------------------------------------------------------------
// fp32 GEMM via V_WMMA_F32_16X16X4_F32.
// One wave (32 lanes) computes one 16x16 tile of C = A(MxK) * B(KxN).
// A operand (16x4 f32): lanes 0-15 carry K={k0,k0+1}, lanes 16-31 K={k0+2,k0+3}
// B operand (4x16 f32): mirrored; C/D: VGPR j holds row M = 8*half + j.
// ---------------------------------------------------------------------------
__global__ void gemm_wmma_f32(const float* __restrict__ A, const float* __restrict__ B,
                              float* __restrict__ C, int M, int K, int Nf) {
    const int m0   = blockIdx.x << 4;
    const int n0   = blockIdx.y << 4;
    const int lane = threadIdx.x;      // 0..31
    const int half = lane >> 4;        // 0 or 1
    const int r    = lane & 15;

    // branchless clamp keeps EXEC all-ones around the WMMA (required by ISA)
    const int rowA = min(m0 + r, M - 1);
    const float* __restrict__ Arow = A + (size_t)rowA * K;
    const int col = n0 + r;

    v8f acc = {};
    for (int k0 = 0; k0 < K; k0 += 4) {
        const int ka = k0 + (half << 1);
        v2f a, b;
        a.x = Arow[ka];
        a.y = Arow[ka + 1];
        b.x = B[(size_t)ka * Nf + col];
        b.y = B[(size_t)(ka + 1) * Nf + col];
        acc = __builtin_amdgcn_wmma_f32_16x16x4_f32(
            /*neg_a=*/false, a, /*neg_b=*/false, b,
            /*c_mod=*/(short)0, acc, /*reuse_a=*/false, /*reuse_b=*/false);
    }

#pragma unroll
    for (int j = 0; j < 8; ++j) {
        const int row = m0 + (half << 3) + j;
        if (row < M) C[(size_t)row * Nf + col] = acc[j];
    }
}

// ---------------------------------------------------------------------------
// Edge scatter: agg[dst] += norm[src] * h[src].  F/4 chunks per edge;
// consecutive threads cover consecutive float4 chunks -> coalesced gathers,
// atomics land in the 192MB L2 (h and agg are L2-resident).
// ---------------------------------------------------------------------------
template <int F>
__global__ void scatter_kernel(const int* __restrict__ src, const int* __restrict__ dst,
                               const float* __restrict__ norm, const float* __restrict__ h,
                               float* __restrict__ agg, long long total) {
    constexpr int CHUNKS = F / 4;  // power of two
    long long t = (long long)blockIdx.x * blockDim.x + threadIdx.x;
    if (t >= total) return;
    const int e = (int)(t / CHUNKS);
    const int c = (int)(t & (CHUNKS - 1));
    const int s = src[e];
    const int d = dst[e];
    const float w = norm[s];
    const float4 hv = *(const float4*)(h + (size_t)s * F + (c << 2));
    float* out = agg + (size_t)d * F + (c << 2);
    atomicAdd(out + 0, hv.x * w);
    atomicAdd(out + 1, hv.y * w);
    atomicAdd(out + 2, hv.z * w);
    atomicAdd(out + 3, hv.w * w);
}

// ---------------------------------------------------------------------------
// out = norm * agg + norm^2 * h + bias  (optional ReLU); elementwise, safe
// in-place when out aliases agg.
// ---------------------------------------------------------------------------
template <int F, bool RELU>
__global__ void combine_kernel(const float* __restrict__ agg, const float* __restrict__ h,
                               const float* __restrict__ norm, const float* __restrict__ bias,
                               float* __restrict__ out, long long total) {
    long long t = (long long)blockIdx.x * blockDim.x + threadIdx.x;
    if (t >= total) return;
    const int i = (int)(t / F);
    const int f = (int)(t & (F - 1));
    const float nm = norm[i];
    float v = nm * agg[t] + (nm * nm) * h[t] + bias[f];
    if (RELU) v = fmaxf(v, 0.0f);
    out[t] = v;
}

// ---------------------------------------------------------------------------
extern "C" void kernel_launch(void* const* d_in, const int* in_sizes, int n_in,
                              void* d_out, int out_size, void* d_ws, size_t ws_size,
                              hipStream_t stream) {
    const float* x    = (const float*)d_in[0];
    const int*   edge = (const int*)  d_in[1];
    const float* W1   = (const float*)d_in[2];
    const float* b1   = (const float*)d_in[3];
    const float* W2   = (const float*)d_in[4];
    const float* b2   = (const float*)d_in[5];

    const int N = in_sizes[0] / F_IN;   // 100000
    const int E = in_sizes[1] / 2;      // 1600000
    const int* src = edge;              // edge_index row 0
    const int* dst = edge + E;          // edge_index row 1

    // Workspace layout (floats): norm[N] | h1[N*64] | agg1/z1[N*64] | h2[N*128]
    float* ws   = (float*)d_ws;
    float* norm = ws;
    float* h1   = ws + (size_t)N;
    float* agg1 = ws + (size_t)N * (1 + F_HID);
    float* h2   = ws + (size_t)N * (1 + 2 * F_HID);
    float* out  = (float*)d_out;        // doubles as agg2

    // 1) zero accumulators (graph-capture-safe)
    hipMemsetAsync(norm, 0, (size_t)N * sizeof(float), stream);
    hipMemsetAsync(agg1, 0, (size_t)N * F_HID * sizeof(float), stream);
    hipMemsetAsync(out,  0, (size_t)N * F_OUT * sizeof(float), stream);

    // 2) degrees + symmetric normalization
    deg_kernel <<<(E + 255) / 256, 256, 0, stream>>>(dst, norm, E);
    norm_kernel<<<(N + 255) / 256, 256, 0, stream>>>(norm, N);

    // 3) h1 = x @ W1  (fp32 WMMA)
    {
        dim3 grid((N + 15) / 16, F_HID / 16);
        gemm_wmma_f32<<<grid, 32, 0, stream>>>(x, W1, h1, N, F_IN, F_HID);
    }

    // 4) agg1[dst] += norm[src] * h1[src]
    {
        long long total = (long long)E * (F_HID / 4);
        scatter_kernel<F_HID><<<(unsigned)((total + 255) / 256), 256, 0, stream>>>(
            src, dst, norm, h1, agg1, total);
    }

    // 5) z1 = relu(norm*agg1 + norm^2*h1 + b1)  (in-place into agg1)
    {
        long long total = (long long)N * F_HID;
        combine_kernel<F_HID, true><<<(unsigned)((total + 255) / 256), 256, 0, stream>>>(
            agg1, h1, norm, b1, agg1, total);
    }

    // 6) h2 = z1 @ W2  (fp32 WMMA)
    {
        dim3 grid((N + 15) / 16, F_OUT / 16);
        gemm_wmma_f32<<<grid, 32, 0, stream>>>(agg1, W2, h2, N, F_HID, F_OUT);
    }

    // 7) out[dst] += norm[src] * h2[src]
    {
        long long total = (long long)E * (F_OUT / 4);
        scatter_kernel<F_OUT><<<(unsigned)((total + 255) / 256), 256, 0, stream>>>(
            src, dst, norm, h2, out, total);
    }

    // 8) out = norm*out + norm^2*h2 + b2   (in-place, no ReLU)
    {
        long long total = (long long)N * F_OUT;
        combine_kernel<F_OUT, false><<<(unsigned)((total + 255) / 256), 256, 0, stream>>>(
            out, h2, norm, b2, out, total);
    }
}